// FALR1_35794257445087
// MI455X (gfx1250) — compile-verified
//
#include <hip/hip_runtime.h>
#include <hip/hip_bf16.h>
#include <math.h>

// ---------------------------------------------------------------------------
// PGN triplet-message GNN layer for MI455X (gfx1250, wave32, WMMA).
// Dominant GEMMs run on V_WMMA_F32_16X16X4_F32 (full f32 precision; the op is
// HBM-bound at ~64MB edge traffic, so f32 matrix pipes are more than enough).
// ---------------------------------------------------------------------------

typedef float v2f __attribute__((ext_vector_type(2)));
typedef float v8f __attribute__((ext_vector_type(8)));

#define BIGN 1000000.0f

constexpr int BB = 8, NN = 128, MD = 128, OD = 128, TRI = 8;
constexpr int ZC = 544;                       // packed z-projection columns:
// [0,128) msg1 | [128,256) msg2 | [256,384) o1 | [384,512) g1
// [512,520) tri1 | [520,528) tri2 | [528,536) tri3 | [536,544) pad

constexpr int ZOUT_N   = BB * NN * ZC;        // 557056
constexpr int MSGG_OFF = ZOUT_N;              // B*128
constexpr int CBF_OFF  = MSGG_OFF + BB * MD;  // B*8   (mean(tri1)+tri_g)
constexpr int E1M_OFF  = CBF_OFF + BB * TRI;  // B*N*8
constexpr int E2M_OFF  = E1M_OFF + BB * NN * TRI;
constexpr int MRED_OFF = E2M_OFF + BB * NN * TRI;  // B*N*128

__device__ inline float zbias(int c,
    const float* m1b, const float* m2b, const float* o1b, const float* g1b,
    const float* t1b, const float* t2b, const float* t3b) {
  if (c < 128) return m1b[c];
  if (c < 256) return m2b[c - 128];
  if (c < 384) return o1b[c - 256];
  if (c < 512) return g1b[c - 384];
  if (c < 520) return t1b[c - 512];
  if (c < 528) return t2b[c - 520];
  if (c < 536) return t3b[c - 528];
  return 0.0f;
}

// --------------------------- K1: z projections -----------------------------
// zout[row, 544] = concat(node,hidden)[row, 256] @ packed W + packed bias
// grid (64 row-tiles, 17 col-groups of 32); block = 64 (2 waves, 1 tile each)
__global__ __launch_bounds__(64) void pgn_zproj(
    const float* __restrict__ node, const float* __restrict__ hidden,
    const float* __restrict__ m1W, const float* __restrict__ m2W,
    const float* __restrict__ o1W, const float* __restrict__ g1W,
    const float* __restrict__ t1W, const float* __restrict__ t2W,
    const float* __restrict__ t3W,
    const float* __restrict__ m1b, const float* __restrict__ m2b,
    const float* __restrict__ o1b, const float* __restrict__ g1b,
    const float* __restrict__ t1b, const float* __restrict__ t2b,
    const float* __restrict__ t3b,
    float* __restrict__ zout) {
  __shared__ float sZ[16 * 260];   // A tile, padded stride for bank-free reads
  __shared__ float sW[32 * 260];   // B tile as [col][k]
  const int rt = blockIdx.x, cg = blockIdx.y;
  const int t = threadIdx.x;
  const int w = t >> 5, lane = t & 31, n = lane & 15, g = lane >> 4;

  {  // stage 16x256 Z tile (coalesced b128 loads)
    const int r = t >> 2, quarter = t & 3;
    const int row = rt * 16 + r;
    for (int i = 0; i < 16; ++i) {
      const int k = quarter * 64 + i * 4;
      const float* src = (k < 128) ? (node + (size_t)row * 128 + k)
                                   : (hidden + (size_t)row * 128 + (k - 128));
      const float4 v4 = *reinterpret_cast<const float4*>(src);
      float* d = sZ + r * 260 + k;
      d[0] = v4.x; d[1] = v4.y; d[2] = v4.z; d[3] = v4.w;
    }
  }
  {  // stage 32x256 weight tile transposed into [c][k]
    const int c2 = t >> 1, half = t & 1;
    const int cglob = cg * 32 + c2;
    for (int i = 0; i < 32; ++i) {
      const int k = half * 128 + i * 4;
      for (int j = 0; j < 4; ++j) {
        float val;
        if (cg < 16) {
          const float* W; int lc;
          if (cglob < 128)      { W = m1W; lc = cglob; }
          else if (cglob < 256) { W = m2W; lc = cglob - 128; }
          else if (cglob < 384) { W = o1W; lc = cglob - 256; }
          else                  { W = g1W; lc = cglob - 384; }
          val = W[(size_t)(k + j) * 128 + lc];
        } else {
          const int c = cglob - 512;
          if (c < 8)       val = t1W[(k + j) * 8 + c];
          else if (c < 16) val = t2W[(k + j) * 8 + (c - 8)];
          else if (c < 24) val = t3W[(k + j) * 8 + (c - 16)];
          else             val = 0.0f;
        }
        sW[c2 * 260 + k + j] = val;
      }
    }
  }
  __syncthreads();

  v8f acc;
#pragma unroll
  for (int v = 0; v < 8; ++v) acc[v] = 0.0f;
  const int c0 = w * 16;
#pragma unroll
  for (int k0 = 0; k0 < 256; k0 += 4) {
    v2f a, bf;
    a.x  = sZ[n * 260 + k0 + 2 * g];        a.y  = sZ[n * 260 + k0 + 2 * g + 1];
    bf.x = sW[(c0 + n) * 260 + k0 + 2 * g]; bf.y = sW[(c0 + n) * 260 + k0 + 2 * g + 1];
    acc = __builtin_amdgcn_wmma_f32_16x16x4_f32(false, a, false, bf,
                                                (short)0, acc, false, false);
  }
  const int colg = cg * 32 + c0 + n;
  const float bias = zbias(colg, m1b, m2b, o1b, g1b, t1b, t2b, t3b);
#pragma unroll
  for (int v = 0; v < 8; ++v) {
    const int row = rt * 16 + v + 8 * g;
    zout[(size_t)row * ZC + colg] = acc[v] + bias;
  }
}

// ------------------ K2: graph projections + mean(tri1) ---------------------
__global__ __launch_bounds__(128) void pgn_graph(
    const float* __restrict__ graph,
    const float* __restrict__ mgW, const float* __restrict__ mgb,
    const float* __restrict__ tgW, const float* __restrict__ tgb,
    const float* __restrict__ zout,
    float* __restrict__ msgg, float* __restrict__ cbf) {
  const int b = blockIdx.x, t = threadIdx.x;
  __shared__ float sg[128];
  sg[t] = graph[b * 128 + t];
  __syncthreads();
  float acc = mgb[t];
  for (int k = 0; k < 128; ++k) acc += sg[k] * mgW[k * 128 + t];
  msgg[b * 128 + t] = acc;
  if (t < 8) {
    float tg = tgb[t];
    for (int k = 0; k < 128; ++k) tg += sg[k] * tgW[k * 8 + t];
    float s = 0.0f;
    for (int nn = 0; nn < 128; ++nn)
      s += zout[(size_t)(b * 128 + nn) * ZC + 512 + t];
    cbf[b * 8 + t] = tg + s * (1.0f / 128.0f);
  }
}

// --------- K3: edge GEMM + masked max over i + te means + te3 --------------
// grid (q-tile 8, b 8, seg 5); block 128 = 4 waves.
// seg<4: cols 32*seg..+31 of me.W; seg==4: [te1|te2|te3|pad] (32 cols).
// Each wave owns a private 16x128 LDS edge tile and p in {w, w+4, ...}, so
// the i-reduction (max / mean) is fully block-local (no global atomics).
__global__ __launch_bounds__(128) void pgn_edge(
    const float* __restrict__ edge, const int* __restrict__ adj,
    const float* __restrict__ meW, const float* __restrict__ meb,
    const float* __restrict__ te1W, const float* __restrict__ te1b,
    const float* __restrict__ te2W, const float* __restrict__ te2b,
    const float* __restrict__ te3W, const float* __restrict__ te3b,
    const float* __restrict__ zout, const float* __restrict__ msgg,
    float* __restrict__ mred, float* __restrict__ e1m, float* __restrict__ e2m,
    float* __restrict__ outtri) {
  __shared__ float sE[4 * 16 * 132];  // 4 per-wave A tiles (also reduce buf)
  __shared__ float sW[32 * 132];      // B tile as [col][k]
  __shared__ int sVm[8];
  const int q0 = blockIdx.x * 16, b = blockIdx.y, z = blockIdx.z;
  const int t = threadIdx.x, w = t >> 5, lane = t & 31, n = lane & 15, g = lane >> 4;

  {  // stage 32-col weight segment transposed
    const int k = t;
    if (z < 4) {
      const float* src = meW + (size_t)k * 128 + z * 32;
      for (int c = 0; c < 32; ++c) sW[c * 132 + k] = src[c];
    } else {
      for (int c = 0; c < 32; ++c) {
        float val = 0.0f;
        if (c < 8)       val = te1W[k * 8 + c];
        else if (c < 16) val = te2W[k * 8 + (c - 8)];
        else if (c < 24) val = te3W[k * 8 + (c - 16)];
        sW[c * 132 + k] = val;
      }
    }
  }
  float bias0, bias1;
  if (z < 4) { bias0 = meb[z * 32 + n]; bias1 = meb[z * 32 + 16 + n]; }
  else       { bias0 = (n < 8) ? te1b[n] : te2b[n - 8];
               bias1 = (n < 8) ? te3b[n] : 0.0f; }

  float* myE = sE + w * (16 * 132);
  v8f rmax0, rmax1, ssum;
#pragma unroll
  for (int i = 0; i < 8; ++i) { rmax0[i] = -BIGN; rmax1[i] = -BIGN; ssum[i] = 0.0f; }
  int vmask = 0;
  __syncthreads();

  for (int iter = 0; iter < 32; ++iter) {
    const int p = w + 4 * iter;
    const float* erow = edge + ((size_t)((b * 128 + p) * 128) + q0) * 128;
    for (int r = 0; r < 16; ++r) {
      const float4 v4 = *reinterpret_cast<const float4*>(erow + r * 128 + lane * 4);
      float* d = myE + r * 132 + lane * 4;
      d[0] = v4.x; d[1] = v4.y; d[2] = v4.z; d[3] = v4.w;
    }
    if (iter < 31)  // CDNA5 global_prefetch_b8 for next p's edge rows
      __builtin_prefetch(edge + ((size_t)((b * 128 + p + 4) * 128) + q0) * 128 + lane * 64, 0, 1);
    __syncthreads();

    v8f acc0, acc1;
#pragma unroll
    for (int i = 0; i < 8; ++i) { acc0[i] = 0.0f; acc1[i] = 0.0f; }
#pragma unroll
    for (int k0 = 0; k0 < 128; k0 += 4) {
      v2f a, b0f, b1f;
      a.x   = myE[n * 132 + k0 + 2 * g];        a.y   = myE[n * 132 + k0 + 2 * g + 1];
      b0f.x = sW[n * 132 + k0 + 2 * g];         b0f.y = sW[n * 132 + k0 + 2 * g + 1];
      b1f.x = sW[(16 + n) * 132 + k0 + 2 * g];  b1f.y = sW[(16 + n) * 132 + k0 + 2 * g + 1];
      acc0 = __builtin_amdgcn_wmma_f32_16x16x4_f32(false, a, false, b0f, (short)0, acc0, false, false);
      acc1 = __builtin_amdgcn_wmma_f32_16x16x4_f32(false, a, false, b1f, (short)0, acc1, false, false);
    }
    __syncthreads();

    if (z < 4) {
      const size_t zr = (size_t)(b * 128 + p) * ZC;
      const float m2a  = zout[zr + 128 + z * 32 + n];
      const float m2bv = zout[zr + 128 + z * 32 + 16 + n];
      const int* arow = adj + (size_t)(b * 128 + p) * 128 + q0;
#pragma unroll
      for (int v = 0; v < 8; ++v) {
        if (arow[v + 8 * g] != 0) {
          vmask |= (1 << v);
          rmax0[v] = fmaxf(rmax0[v], acc0[v] + bias0 + m2a);
          rmax1[v] = fmaxf(rmax1[v], acc1[v] + bias1 + m2bv);
        }
      }
    } else {
#pragma unroll
      for (int v = 0; v < 8; ++v) {
        ssum[v] += acc0[v] + bias0;                    // te1|te2 column sums
        if (n < 8) {                                   // te3 -> tri region
          const int q = q0 + v + 8 * g;
          outtri[((size_t)(b * 128 + p) * 128 + q) * 8 + n] = acc1[v] + bias1;
        }
      }
    }
  }
  __syncthreads();

  if (z < 4) {  // cross-wave max + validity, then finalize msgs_red
    float* red = sE;
#pragma unroll
    for (int v = 0; v < 8; ++v) {
      red[w * 512 + v * 32 + lane]       = rmax0[v];
      red[w * 512 + 256 + v * 32 + lane] = rmax1[v];
    }
    if (n == 0) sVm[w * 2 + g] = vmask;
    __syncthreads();
    for (int i = 0; i < 4; ++i) {
      const int idx = t + 128 * i;                 // 16 q x 32 c
      const int q = idx >> 5, c = idx & 31;
      const int tile = c >> 4, nn = c & 15, gg = q >> 3, vv = q & 7;
      const int ln = gg * 16 + nn;
      float mv = -BIGN; int anyv = 0;
      for (int ww = 0; ww < 4; ++ww) {
        mv = fmaxf(mv, red[ww * 512 + tile * 256 + vv * 32 + ln]);
        anyv |= (sVm[ww * 2 + gg] >> vv) & 1;
      }
      const int m = z * 32 + c;
      float outv = -BIGN;
      if (anyv)
        outv = mv + zout[(size_t)(b * 128 + q0 + q) * ZC + m] + msgg[b * 128 + m];
      mred[(size_t)(b * 128 + q0 + q) * 128 + m] = outv;
    }
  } else {      // cross-wave sum -> e1m / e2m column means
    float* red = sE;
#pragma unroll
    for (int v = 0; v < 8; ++v) red[w * 256 + v * 32 + lane] = ssum[v];
    __syncthreads();
    for (int i = 0; i < 2; ++i) {
      const int idx = t + 128 * i;                 // 16 q x 16 c
      const int q = idx >> 4, nn = idx & 15;
      const int gg = q >> 3, vv = q & 7, ln = gg * 16 + nn;
      float s = 0.0f;
      for (int ww = 0; ww < 4; ++ww) s += red[ww * 256 + vv * 32 + ln];
      s *= (1.0f / 128.0f);
      if (nn < 8) e1m[(size_t)(b * 128 + q0 + q) * 8 + nn] = s;
      else        e2m[(size_t)(b * 128 + q0 + q) * 8 + (nn - 8)] = s;
    }
  }
}

// --------------------- K4: output head + gating ----------------------------
__global__ __launch_bounds__(256) void pgn_out(
    const float* __restrict__ mred, const float* __restrict__ zout,
    const float* __restrict__ hidden,
    const float* __restrict__ o2W, const float* __restrict__ o2b,
    const float* __restrict__ g2W, const float* __restrict__ g2b,
    const float* __restrict__ g3W, const float* __restrict__ g3b,
    float* __restrict__ out) {
  const int row = blockIdx.x, t = threadIdx.x;
  __shared__ float sr[128];
  __shared__ float sg[128];
  if (t < 128) sr[t] = mred[(size_t)row * 128 + t];
  __syncthreads();
  float s1;
  if (t < 128) {
    s1 = o2b[t];
    for (int k = 0; k < 128; ++k) s1 += sr[k] * o2W[k * 128 + t];
  } else {
    const int c = t - 128;
    s1 = g2b[c];
    for (int k = 0; k < 128; ++k) s1 += sr[k] * g2W[k * 128 + c];
    sg[c] = fmaxf(0.0f, zout[(size_t)row * ZC + 384 + c] + s1);
  }
  __syncthreads();
  if (t < 128) {
    float gd = g3b[t];
    for (int k = 0; k < 128; ++k) gd += sg[k] * g3W[k * 128 + t];
    const float gate = 1.0f / (1.0f + __expf(-gd));
    const float ret = fmaxf(0.0f, zout[(size_t)row * ZC + 256 + t] + s1);
    const float h = hidden[(size_t)row * 128 + t];
    out[(size_t)row * 128 + t] = ret * gate + h * (1.0f - gate);
  }
}

// ------------------- K5: finalize tri_msgs in place ------------------------
__global__ __launch_bounds__(256) void pgn_tri(
    const float* __restrict__ zout, const float* __restrict__ cbf,
    const float* __restrict__ e1m, const float* __restrict__ e2m,
    float* __restrict__ outtri) {
  const int row = blockIdx.x;            // row = b*128 + j
  const int b = row >> 7;
  const int t = threadIdx.x;
  __shared__ float radd[8];
  __shared__ float cadd[1024];
  if (t < 8)
    radd[t] = cbf[b * 8 + t] + zout[(size_t)row * ZC + 520 + t]
            + e1m[(size_t)row * 8 + t];
  for (int i = 0; i < 4; ++i) {
    const int idx = t + 256 * i;
    const int k = idx >> 3, f = idx & 7;
    cadd[idx] = zout[(size_t)(b * 128 + k) * ZC + 528 + f]
              + e2m[(size_t)(b * 128 + k) * 8 + f];
  }
  __syncthreads();
  float* base = outtri + (size_t)row * 1024;
  for (int i = 0; i < 4; ++i) {
    const int idx = t + 256 * i;
    const int f = idx & 7;
    base[idx] = fmaxf(0.0f, base[idx] + radd[f] + cadd[idx]);
  }
}

// ---------------------------------------------------------------------------
extern "C" void kernel_launch(void* const* d_in, const int* in_sizes, int n_in,
                              void* d_out, int out_size, void* d_ws, size_t ws_size,
                              hipStream_t stream) {
  (void)in_sizes; (void)n_in; (void)out_size; (void)ws_size;
  const float* node   = (const float*)d_in[0];
  const float* edge   = (const float*)d_in[1];
  const float* graph  = (const float*)d_in[2];
  const int*   adj    = (const int*)d_in[3];
  const float* hidden = (const float*)d_in[4];
  // params flattened in setup_inputs() insertion order, {W, b} per entry
  const float* m1W  = (const float*)d_in[5];  const float* m1b  = (const float*)d_in[6];
  const float* m2W  = (const float*)d_in[7];  const float* m2b  = (const float*)d_in[8];
  const float* meW  = (const float*)d_in[9];  const float* meb  = (const float*)d_in[10];
  const float* mgW  = (const float*)d_in[11]; const float* mgb  = (const float*)d_in[12];
  const float* o1W  = (const float*)d_in[13]; const float* o1b  = (const float*)d_in[14];
  const float* o2W  = (const float*)d_in[15]; const float* o2b  = (const float*)d_in[16];
  const float* t1W  = (const float*)d_in[17]; const float* t1b  = (const float*)d_in[18];
  const float* t2W  = (const float*)d_in[19]; const float* t2b  = (const float*)d_in[20];
  const float* t3W  = (const float*)d_in[21]; const float* t3b  = (const float*)d_in[22];
  const float* te1W = (const float*)d_in[23]; const float* te1b = (const float*)d_in[24];
  const float* te2W = (const float*)d_in[25]; const float* te2b = (const float*)d_in[26];
  const float* te3W = (const float*)d_in[27]; const float* te3b = (const float*)d_in[28];
  const float* tgW  = (const float*)d_in[29]; const float* tgb  = (const float*)d_in[30];
  const float* g1W  = (const float*)d_in[31]; const float* g1b  = (const float*)d_in[32];
  const float* g2W  = (const float*)d_in[33]; const float* g2b  = (const float*)d_in[34];
  const float* g3W  = (const float*)d_in[35]; const float* g3b  = (const float*)d_in[36];

  float* ws   = (float*)d_ws;
  float* zout = ws;
  float* msgg = ws + MSGG_OFF;
  float* cbf  = ws + CBF_OFF;
  float* e1m  = ws + E1M_OFF;
  float* e2m  = ws + E2M_OFF;
  float* mred = ws + MRED_OFF;
  float* outr   = (float*)d_out;
  float* outtri = outr + BB * NN * OD;   // tri_msgs region (B,N,N,8)

  pgn_zproj<<<dim3(64, 17), 64, 0, stream>>>(
      node, hidden, m1W, m2W, o1W, g1W, t1W, t2W, t3W,
      m1b, m2b, o1b, g1b, t1b, t2b, t3b, zout);
  pgn_graph<<<8, 128, 0, stream>>>(graph, mgW, mgb, tgW, tgb, zout, msgg, cbf);
  pgn_edge<<<dim3(8, 8, 5), 128, 0, stream>>>(
      edge, adj, meW, meb, te1W, te1b, te2W, te2b, te3W, te3b,
      zout, msgg, mred, e1m, e2m, outtri);
  pgn_out<<<1024, 256, 0, stream>>>(mred, zout, hidden,
                                    o2W, o2b, g2W, g2b, g3W, g3b, outr);
  pgn_tri<<<1024, 256, 0, stream>>>(zout, cbf, e1m, e2m, outtri);
}